// Nat2D_21921513079539
// MI455X (gfx1250) — compile-verified
//
#include <hip/hip_runtime.h>
#include <hip/hip_bf16.h>
#include <stdint.h>

// ---------------------------------------------------------------------------
// NAT2D (neighborhood attention), B=1, H=W=64, C=256, NH=8, hd=32, K=7, fp32.
//  ws layout: [0)            qkv   : 4096*768 f32 (12.6 MB)
//             [4096*768)     attno : 4096*256 f32 ( 4.2 MB)
// ---------------------------------------------------------------------------

typedef float v2f __attribute__((ext_vector_type(2)));
typedef float v8f __attribute__((ext_vector_type(8)));

#define HH 64
#define WW 64
#define CC 256
#define NHEAD 8
#define HD 32
#define KW 7          // window size
#define RSPAN 14      // max halo rows/cols for an 8x8 query tile: 8 + (7-1)

// ---------------------------------------------------------------------------
// Kernels 1/3: fp32 WMMA GEMM.  C[m,n] = sum_k A[m,k] * B[n,k] + bias[n]
// One wave computes one 32x32 tile: 2 A-frags x 2 B-frags -> 4 WMMAs per
// K-step (4 global_load_b64 : 4 v_wmma, 2x the reuse of a 16x16 tile).
// Fragment layout (ISA 16x4 / 4x16 fp32): lane m/n = lane&15; lanes 0-15
// hold K={0,1}, lanes 16-31 hold K={2,3} -> one aligned float2 per lane.
// ---------------------------------------------------------------------------
__global__ __launch_bounds__(256)
void wmma_gemm_bias_f32(const float* __restrict__ A,
                        const float* __restrict__ B,
                        const float* __restrict__ bias,
                        float* __restrict__ C,
                        int M, int N, int Kdim,
                        int lda, int ldb, int ldc) {
    const int lane = threadIdx.x & 31;
    const int wave = threadIdx.x >> 5;
    const int tiles_n = N >> 5;
    const int tile = blockIdx.x * (blockDim.x >> 5) + wave;
    if (tile >= (M >> 5) * tiles_n) return;
    const int tm = (tile / tiles_n) << 5;
    const int tn = (tile % tiles_n) << 5;

    const int half = lane >> 4;   // 0: lanes 0-15 (K 0,1) | 1: lanes 16-31 (K 2,3)
    const int sub  = lane & 15;   // row (A) / col (B) within 16-wide subtile

    const float* __restrict__ Arow0 = A + (size_t)(tm + sub) * lda + 2 * half;
    const float* __restrict__ Arow1 = Arow0 + (size_t)16 * lda;
    const float* __restrict__ Brow0 = B + (size_t)(tn + sub) * ldb + 2 * half;
    const float* __restrict__ Brow1 = Brow0 + (size_t)16 * ldb;

    v8f acc00 = {}, acc01 = {}, acc10 = {}, acc11 = {};
    for (int k = 0; k < Kdim; k += 4) {
        v2f a0 = *(const v2f*)(Arow0 + k);   // global_load_b64, 8B aligned
        v2f a1 = *(const v2f*)(Arow1 + k);
        v2f b0 = *(const v2f*)(Brow0 + k);
        v2f b1 = *(const v2f*)(Brow1 + k);
        acc00 = __builtin_amdgcn_wmma_f32_16x16x4_f32(false, a0, false, b0,
                                                      (short)0, acc00, false, false);
        acc01 = __builtin_amdgcn_wmma_f32_16x16x4_f32(false, a0, false, b1,
                                                      (short)0, acc01, false, false);
        acc10 = __builtin_amdgcn_wmma_f32_16x16x4_f32(false, a1, false, b0,
                                                      (short)0, acc10, false, false);
        acc11 = __builtin_amdgcn_wmma_f32_16x16x4_f32(false, a1, false, b1,
                                                      (short)0, acc11, false, false);
    }

    // C/D layout: lane covers N = sub, M = vgpr + 8*half (per 16x16 subtile)
    const float bn0 = bias[tn + sub];
    const float bn1 = bias[tn + 16 + sub];
    float* __restrict__ Crow = C + (size_t)(tm + 8 * half) * ldc + tn + sub;
#pragma unroll
    for (int v = 0; v < 8; ++v) {
        Crow[(size_t)v * ldc]                   = acc00[v] + bn0;
        Crow[(size_t)v * ldc + 16]              = acc01[v] + bn1;
        Crow[(size_t)(v + 16) * ldc]            = acc10[v] + bn0;
        Crow[(size_t)(v + 16) * ldc + 16]       = acc11[v] + bn1;
    }
}

// ---------------------------------------------------------------------------
// Kernel 2: neighborhood attention.
// grid = (W/8, H/8, NHEAD), block = (8,8).  One thread = one query pixel.
// K/V halo (<=14x14x32 f32 each, 50 KB total) staged in LDS via the gfx1250
// async memory->LDS path (GLOBAL_LOAD_ASYNC_TO_LDS_B128, ASYNCcnt-tracked).
// ---------------------------------------------------------------------------
__global__ __launch_bounds__(64)
void nat2d_attn(const float* __restrict__ qkv,   // [4096, 768]
                const float* __restrict__ rpb,   // [8, 13, 13]
                float* __restrict__ attno)       // [4096, 256]
{
    __shared__ float kbuf[RSPAN * RSPAN * HD];
    __shared__ float vbuf[RSPAN * RSPAN * HD];

    const int head = blockIdx.z;
    const int i0 = blockIdx.y * 8;
    const int j0 = blockIdx.x * 8;
    const int tid = threadIdx.y * 8 + threadIdx.x;

    // union of clamped windows for the 8 queries in each dimension
    const int r0 = min(max(i0 - 3, 0), HH - KW);
    const int rl = min(i0 + 4, HH - KW) + (KW - 1);
    const int c0 = min(max(j0 - 3, 0), WW - KW);
    const int cl = min(j0 + 4, WW - KW) + (KW - 1);
    const int nr = rl - r0 + 1;     // <= 14
    const int nc = cl - c0 + 1;     // <= 14

    // LDS byte offsets (generic shared ptr low 32 bits == LDS offset,
    // per the shared aperture mapping)
    const uint32_t kbase = (uint32_t)(size_t)kbuf;
    const uint32_t vbase = (uint32_t)(size_t)vbuf;

    // cooperative halo load, 16B per lane per issue, direct to LDS
    const int n4 = nr * nc * (HD / 4);
    for (int idx = tid; idx < n4; idx += 64) {
        const int d4 = idx & 7;
        const int rc = idx >> 3;
        const int c  = rc % nc;
        const int r  = rc / nc;
        const uint64_t src = (uint64_t)(const void*)(
            qkv + (size_t)((r0 + r) * WW + (c0 + c)) * (3 * CC)
                + CC + head * HD + d4 * 4);
        const uint32_t dst = (uint32_t)(((r * nc + c) * HD + d4 * 4) * sizeof(float));
        asm volatile("global_load_async_to_lds_b128 %0, %1, off"
                     :: "v"(kbase + dst), "v"(src) : "memory");
        asm volatile("global_load_async_to_lds_b128 %0, %1, off"
                     :: "v"(vbase + dst), "v"(src + CC * sizeof(float)) : "memory");
    }
    asm volatile("s_wait_asynccnt 0x0" ::: "memory");
    __syncthreads();

    const int i = i0 + (int)threadIdx.y;
    const int j = j0 + (int)threadIdx.x;
    const int si = min(max(i - 3, 0), HH - KW);   // window start (clamped)
    const int sj = min(max(j - 3, 0), WW - KW);

    // q * scale
    const float scale = 0.17677669529663687f;     // 32^-0.5
    const float* qp = qkv + (size_t)(i * WW + j) * (3 * CC) + head * HD;
    float4 q[HD / 4];
#pragma unroll
    for (int d = 0; d < HD / 4; ++d) {
        float4 t = ((const float4*)qp)[d];
        q[d] = make_float4(t.x * scale, t.y * scale, t.z * scale, t.w * scale);
    }

    // logits + relative position bias
    float logit[KW * KW];
    const float* rp = rpb + head * (2 * KW - 1) * (2 * KW - 1);
#pragma unroll
    for (int a = 0; a < KW; ++a) {
        const int kr = si + a - r0;
        const int bi = si + a - i + (KW - 1);
#pragma unroll
        for (int b = 0; b < KW; ++b) {
            const int kc = sj + b - c0;
            const float4* kp = (const float4*)&kbuf[(kr * nc + kc) * HD];
            float acc = 0.f;
#pragma unroll
            for (int d = 0; d < HD / 4; ++d) {
                float4 kd = kp[d];
                acc += q[d].x * kd.x + q[d].y * kd.y + q[d].z * kd.z + q[d].w * kd.w;
            }
            const int bj = sj + b - j + (KW - 1);
            logit[a * KW + b] = acc + rp[bi * (2 * KW - 1) + bj];
        }
    }

    // softmax over 49
    float m = logit[0];
#pragma unroll
    for (int n = 1; n < KW * KW; ++n) m = fmaxf(m, logit[n]);
    float s = 0.f;
#pragma unroll
    for (int n = 0; n < KW * KW; ++n) { logit[n] = __expf(logit[n] - m); s += logit[n]; }
    const float inv = 1.f / s;

    // weighted sum of V
    float4 out[HD / 4];
#pragma unroll
    for (int d = 0; d < HD / 4; ++d) out[d] = make_float4(0.f, 0.f, 0.f, 0.f);
#pragma unroll
    for (int a = 0; a < KW; ++a) {
        const int kr = si + a - r0;
#pragma unroll
        for (int b = 0; b < KW; ++b) {
            const int kc = sj + b - c0;
            const float w = logit[a * KW + b] * inv;
            const float4* vp = (const float4*)&vbuf[(kr * nc + kc) * HD];
#pragma unroll
            for (int d = 0; d < HD / 4; ++d) {
                float4 vd = vp[d];
                out[d].x += w * vd.x; out[d].y += w * vd.y;
                out[d].z += w * vd.z; out[d].w += w * vd.w;
            }
        }
    }

    float4* op = (float4*)(attno + (size_t)(i * WW + j) * CC + head * HD);
#pragma unroll
    for (int d = 0; d < HD / 4; ++d) op[d] = out[d];
}

// ---------------------------------------------------------------------------
extern "C" void kernel_launch(void* const* d_in, const int* in_sizes, int n_in,
                              void* d_out, int out_size, void* d_ws, size_t ws_size,
                              hipStream_t stream) {
    (void)in_sizes; (void)n_in; (void)out_size; (void)ws_size;
    const float* x      = (const float*)d_in[0];   // [64,64,256]
    const float* qkv_w  = (const float*)d_in[1];   // [768,256]
    const float* qkv_b  = (const float*)d_in[2];   // [768]
    const float* proj_w = (const float*)d_in[3];   // [256,256]
    const float* proj_b = (const float*)d_in[4];   // [256]
    const float* rpb    = (const float*)d_in[5];   // [8,13,13]
    float* out = (float*)d_out;                    // [64,64,256]

    float* qkv   = (float*)d_ws;                   // [4096,768]
    float* attno = qkv + (size_t)4096 * 768;       // [4096,256]

    // 1) QKV projection: [4096,256] x [768,256]^T -> [4096,768]
    {
        const int M = 4096, N = 768, K = 256;
        const int tiles = (M / 32) * (N / 32);     // 3072 waves
        wmma_gemm_bias_f32<<<tiles / 8, 256, 0, stream>>>(
            x, qkv_w, qkv_b, qkv, M, N, K, 256, 256, 768);
    }

    // 2) neighborhood attention
    {
        dim3 grid(WW / 8, HH / 8, NHEAD);
        dim3 block(8, 8);
        nat2d_attn<<<grid, block, 0, stream>>>(qkv, rpb, attno);
    }

    // 3) output projection: [4096,256] x [256,256]^T -> d_out
    {
        const int M = 4096, N = 256, K = 256;
        const int tiles = (M / 32) * (N / 32);     // 1024 waves
        wmma_gemm_bias_f32<<<tiles / 8, 256, 0, stream>>>(
            attno, proj_w, proj_b, out, M, N, K, 256, 256, 256);
    }
}